// opool_37048387895526
// MI455X (gfx1250) — compile-verified
//
#include <hip/hip_runtime.h>
#include <stdint.h>

#define D_OR   12
#define NB     8
#define NC     64
#define HW     16384                 // 128*128
#define SLICES (NB * NC * D_OR)      // 6144 per-(b,c,d) slices
#define GROUPS (NB * NC)             // 512 (b,c) groups
#define TILE_F4 1024                 // 4096 floats = 16 KB LDS tile
#define TILES_PER_GROUP 4            // 4 * 4096 = 16384 floats per group

// ---------- CDNA5 async global<->LDS helpers (inline asm; toolchain-portable) ----------
__device__ __forceinline__ unsigned lds_byte_off(const void* p) {
  // generic -> LDS addrspace(3) -> 32-bit byte offset from LDS base
  return (unsigned)(unsigned long long)(const __attribute__((address_space(3))) void*)p;
}

__device__ __forceinline__ void async_load_g2l_b128(unsigned lds_off, const void* gaddr) {
  // VDST = LDS byte-offset VGPR, VADDR = 64-bit global address, saddr = off
  asm volatile("global_load_async_to_lds_b128 %0, %1, off"
               :
               : "v"(lds_off), "v"((unsigned long long)gaddr)
               : "memory");
}

__device__ __forceinline__ void async_store_l2g_b128(void* gaddr, unsigned lds_off) {
  // VADDR = 64-bit global address, VSRC = LDS byte-offset VGPR
  asm volatile("global_store_async_from_lds_b128 %0, %1, off"
               :
               : "v"((unsigned long long)gaddr), "v"(lds_off)
               : "memory");
}

__device__ __forceinline__ void wait_asynccnt0() {
  asm volatile("s_wait_asynccnt 0x0" ::: "memory");
}

// ---------------- Phase 1: per-(b,c,d) slice max (streams full 403 MB input) ----------------
__global__ __launch_bounds__(256) void opool_slice_max_kernel(const float* __restrict__ x,
                                                              float* __restrict__ slice_max) {
  const int g   = blockIdx.x;                 // g = b*768 + c*12 + d
  const int tid = threadIdx.x;
  const float4* p = (const float4*)x + (size_t)g * (HW / 4);

  float m = -__builtin_inff();
#pragma unroll
  for (int j = 0; j < 16; ++j) {              // 16 x b128 per lane, coalesced
    float4 v = p[tid + j * 256];
    m = fmaxf(m, fmaxf(fmaxf(v.x, v.y), fmaxf(v.z, v.w)));
  }

  __shared__ float red[256];
  red[tid] = m;
  __syncthreads();
#pragma unroll
  for (int s = 128; s > 0; s >>= 1) {
    if (tid < s) red[tid] = fmaxf(red[tid], red[tid + s]);
    __syncthreads();
  }
  if (tid == 0) slice_max[g] = red[0];
}

// ------- Phase 2: 12-way argmax (first-occurrence tie-break) + async DMA gather-copy -------
__global__ __launch_bounds__(256) void opool_gather_copy_kernel(const float* __restrict__ x,
                                                                const float* __restrict__ slice_max,
                                                                float* __restrict__ out) {
  const int group = blockIdx.y;               // b*64 + c
  const int tile  = blockIdx.x;               // 0..3
  const int tid   = threadIdx.x;

  __shared__ int   sh_d;
  __shared__ float4 buf[TILE_F4];             // 16 KB staging tile

  if (tid == 0) {
    const float* mx = slice_max + group * D_OR;   // group*12 + d == b*768 + c*12 + d
    float best = mx[0];
    int   bd   = 0;
#pragma unroll
    for (int d = 1; d < D_OR; ++d) {
      float v = mx[d];
      if (v > best) { best = v; bd = d; }     // strict > keeps lowest d on ties
    }
    sh_d = bd;
  }
  __syncthreads();
  const int d = sh_d;

  const size_t src_f4 = (((size_t)group * D_OR + d) * HW + (size_t)tile * (TILE_F4 * 4)) / 4;
  const size_t dst_f4 = (((size_t)group * HW) + (size_t)tile * (TILE_F4 * 4)) / 4;
  const float4* src = (const float4*)x + src_f4;
  float4*       dst = (float4*)out     + dst_f4;

  // global -> LDS via async data mover (no VGPR staging)
#pragma unroll
  for (int j = 0; j < 4; ++j) {
    const int idx = tid + j * 256;
    async_load_g2l_b128(lds_byte_off(&buf[idx]), (const void*)&src[idx]);
  }
  wait_asynccnt0();                           // per-wave: its LDS region is now valid

  // LDS -> global via async data mover
#pragma unroll
  for (int j = 0; j < 4; ++j) {
    const int idx = tid + j * 256;
    async_store_l2g_b128((void*)&dst[idx], lds_byte_off(&buf[idx]));
  }
  wait_asynccnt0();
}

extern "C" void kernel_launch(void* const* d_in, const int* in_sizes, int n_in,
                              void* d_out, int out_size, void* d_ws, size_t ws_size,
                              hipStream_t stream) {
  (void)in_sizes; (void)n_in; (void)out_size; (void)ws_size;
  const float* x   = (const float*)d_in[0];
  float* out       = (float*)d_out;
  float* slice_max = (float*)d_ws;            // 6144 floats = 24 KB scratch

  opool_slice_max_kernel<<<SLICES, 256, 0, stream>>>(x, slice_max);

  dim3 grid(TILES_PER_GROUP, GROUPS);
  opool_gather_copy_kernel<<<grid, dim3(256), 0, stream>>>(x, slice_max, out);
}